// SoftGlidingBoxesLayer_50319836839992
// MI455X (gfx1250) — compile-verified
//
#include <hip/hip_runtime.h>

typedef __attribute__((ext_vector_type(2))) float v2f;
typedef __attribute__((ext_vector_type(4))) float v4f;
typedef __attribute__((ext_vector_type(8))) float v8f;
typedef __attribute__((ext_vector_type(4))) unsigned int u32x4;
typedef __attribute__((ext_vector_type(4))) int i32x4;
typedef __attribute__((ext_vector_type(8))) int i32x8;

#define HH 224
#define WW 224
#define PITCH 228              // multiple of 4 for b128 LDS ops; 224..227 are pad
#define NTHREADS 256
#define NSCALES 10
#define ROWSTRIDE (WW * 64)    // 14336 floats between h rows in global x

__device__ __forceinline__ v4f vmax4(v4f a, v4f b) {
  v4f r;
  r.x = fmaxf(a.x, b.x);
  r.y = fmaxf(a.y, b.y);
  r.z = fmaxf(a.z, b.z);
  r.w = fmaxf(a.w, b.w);
  return r;
}

__global__ __launch_bounds__(NTHREADS)
void sgb_kernel(const float* __restrict__ x, float* __restrict__ out) {
  // Full plane resident in LDS: 224 * 228 * 4 = 204,288 B (CDNA5 320KB WGP LDS)
  __shared__ float plane[HH * PITCH];
  __shared__ float parts[NSCALES][NTHREADS];

  const int tid = threadIdx.x;
  const int bid = blockIdx.x;          // 0..511
  const int dc  = bid & 63;            // d*8 + c  (innermost 64 channels)
  const int b   = bid >> 6;
  const size_t base = (size_t)b * ((size_t)HH * WW * 64) + (size_t)dc;

  // ---- TDM plane fill: ONE tensor_load_to_lds with iterate mode ----
  // Per iteration: tile 1x224, element stride 64*4B (channel gather of one
  // image row, 896B into LDS); iterate 224 times stepping global by one h-row
  // (14336 elements) and LDS by the 228-float pitch. Issued by wave 0 only.
  if (tid < 32) {
    const unsigned long long ga =
        (unsigned long long)(uintptr_t)(x + base);            // byte address
    const unsigned int lds_base = (unsigned int)(uintptr_t)(&plane[0]);

    u32x4 g0;
    g0.x = 1u;                              // count=1 (valid), no gather mode
    g0.y = lds_base;                        // D#.lds_addr (bytes)
    g0.z = (unsigned int)(ga & 0xFFFFFFFFu);         // global_addr[31:0]
    g0.w = (unsigned int)((ga >> 32) & 0x01FFFFFFu)  // global_addr[56:32]
         | 0x80000000u;                              // type=2 ("image")

    i32x8 g1;
    g1[0] = (2 << 16)                        // data_size=4B
          | (1 << 19);                       // iterate_enable
    g1[1] = (1 << 16);                       // tensor_dim0 = 1 (low16)
    g1[2] = (WW << 16);                      // tensor_dim1 = 224 (low16)
    g1[3] = (1 << 16);                       // tile_dim0 = 1
    g1[4] = WW;                              // tile_dim1 = 224, tile_dim2 = 0
    g1[5] = 64;                              // tensor_dim0_stride = 64 elements
    g1[6] = 0;
    g1[7] = 0;

    i32x4 g2;                                // iterate-mode fields
    g2[0] = 0;                               // tensor_dim2 unused
    g2[1] = PITCH;                           // lds_addr_increment (elements)
    g2[2] = ROWSTRIDE;                       // global_addr_increment (elements)
    g2[3] = (HH - 1) << 16;                  // iterate_count = 223 -> 224 iters

    i32x4 g3 = {0, 0, 0, 0};

#if defined(__clang_major__) && (__clang_major__ >= 23)
    i32x8 g4 = {0, 0, 0, 0, 0, 0, 0, 0};
    __builtin_amdgcn_tensor_load_to_lds(g0, g1, g2, g3, g4, 0);
#else
    __builtin_amdgcn_tensor_load_to_lds(g0, g1, g2, g3, 0);
#endif
    __builtin_amdgcn_s_wait_tensorcnt(0);    // wave 0 owns TENSORcnt
  }
  __syncthreads();                           // publish LDS to all waves

  // ---- scale 1: column sums from LDS ----
  float acc1 = 0.f;
  if (tid < WW) {
#pragma unroll 4
    for (int h = 0; h < HH; ++h) acc1 += plane[h * PITCH + tid];
  }
  parts[0][tid] = acc1;
  __syncthreads();

  // ---- scales 2..10: separable in-place 2-tap max sweeps in LDS ----
  for (int s = 2; s <= NSCALES; ++s) {
    const int E = HH + 2 - s;          // valid extent of P_{s-1} on entry

    // Horizontal: P[i][j] = max(P[i][j], P[i][j+1])  (row-private per thread)
    if (tid < E) {
      v4f* row = (v4f*)&plane[tid * PITCH];
      v4f a = row[0];
#pragma unroll 2
      for (int g = 0; g < WW / 4; ++g) {
        v4f nb = row[g + 1];           // g=55 reads pad floats 224..227 (harmless)
        v4f sh;
        sh.x = a.y; sh.y = a.z; sh.z = a.w; sh.w = nb.x;
        row[g] = vmax4(a, sh);
        a = nb;
      }
    }
    __syncthreads();

    // Vertical: P[i][j] = max(P[i][j], P[i+1][j]) + accumulate (column-private)
    float va = 0.f;
    if (tid < E - 1) {
      float prev = plane[tid];
#pragma unroll 4
      for (int i = 0; i < E - 1; ++i) {
        float cur = plane[(i + 1) * PITCH + tid];
        float m = fmaxf(prev, cur);
        plane[i * PITCH + tid] = m;
        va += m;
        prev = cur;
      }
    }
    parts[s - 1][tid] = va;
    __syncthreads();
  }

  // ---- per-scale reduction of 256 partials: WMMA ones-matrix trick ----
  // D = A(16x4, all ones) * B(4x16, partials) + C  =>  every row of D holds
  // the column sums of B; chaining C over 4 blocks sums all 256 partials.
  // Sum of D's VGPR0 over all 32 lanes (rows M=0 and M=8) = 2 * total.
  if (tid < 32) {
    v2f ones;
    ones.x = 1.f; ones.y = 1.f;        // A all-ones => layout-agnostic colsums
    for (int s = 0; s < NSCALES; ++s) {
      v8f d = {};
      for (int i = 0; i < 4; ++i) {
        v2f bb = *(v2f*)&parts[s][i * 64 + tid * 2];  // any bijection into B slots
        d = __builtin_amdgcn_wmma_f32_16x16x4_f32(
            /*neg_a=*/false, ones, /*neg_b=*/false, bb,
            /*c_mod=*/(short)0, d, /*reuse_a=*/false, /*reuse_b=*/false);
      }
      float v = d[0];
      v += __shfl_down(v, 16);
      v += __shfl_down(v, 8);
      v += __shfl_down(v, 4);
      v += __shfl_down(v, 2);
      v += __shfl_down(v, 1);
      if (tid == 0) {
        float total = v * 0.5f;        // rows M=0 and M=8 are duplicates
        out[b * 640 + dc * 10 + s] = fmaxf(total, 0.f) + 1.f;
      }
    }
  }
}

extern "C" void kernel_launch(void* const* d_in, const int* in_sizes, int n_in,
                              void* d_out, int out_size, void* d_ws, size_t ws_size,
                              hipStream_t stream) {
  (void)in_sizes; (void)n_in; (void)d_ws; (void)ws_size; (void)out_size;
  const float* x = (const float*)d_in[0];
  float* out = (float*)d_out;
  // 8 batches * 64 (d,c) channels = 512 planes, one workgroup each (8 waves)
  sgb_kernel<<<dim3(512), dim3(NTHREADS), 0, stream>>>(x, out);
}